// SelfAttention_Gamma_84739704750414
// MI455X (gfx1250) — compile-verified
//
#include <hip/hip_runtime.h>

typedef float v2f __attribute__((ext_vector_type(2)));
typedef float v8f __attribute__((ext_vector_type(8)));

// ---------------------------------------------------------------------------
// Y[m,n] = sum_k A[m,k] * B[n,k]  (+ bias[n])     A:[M,K] B:[N,K] row-major
// grid = (N/256, M/64), block = 256 (8 waves arranged 2(M) x 4(N)).
// Each wave: 32x64 output = 2x4 WMMA tiles (fp32 16x16x4), K double-buffered.
// Per K-step: 6 global_load_b64 feed 8 v_wmma.
// ---------------------------------------------------------------------------
__global__ __launch_bounds__(256) void gemm_abt(const float* __restrict__ A,
                                                const float* __restrict__ B,
                                                float* __restrict__ Y,
                                                int M, int N, int K,
                                                const float* __restrict__ bias)
{
    const int lane = threadIdx.x & 31;
    const int wave = threadIdx.x >> 5;
    const int wm   = wave & 1;             // 2 waves along M
    const int wn   = wave >> 1;            // 4 waves along N
    const int m0   = blockIdx.y * 64  + wm * 32;
    const int n0   = blockIdx.x * 256 + wn * 64;
    const int mr   = lane & 15;            // row-within-tile (A) / col (B,D)
    const int kofs = (lane >> 4) << 1;     // K sub-offset: 0 or 2

    // 32-bit element offsets off uniform bases (all operands < 2^29 elements)
    unsigned aoff[2], boff[4];
    aoff[0] = (unsigned)(m0 + mr) * (unsigned)K + (unsigned)kofs;
    aoff[1] = aoff[0] + 16u * (unsigned)K;
#pragma unroll
    for (int t = 0; t < 4; ++t)
        boff[t] = (unsigned)(n0 + t * 16 + mr) * (unsigned)K + (unsigned)kofs;

    v8f acc[2][4];
#pragma unroll
    for (int i = 0; i < 2; ++i)
#pragma unroll
        for (int t = 0; t < 4; ++t) acc[i][t] = (v8f){};

    // prologue: fragments for k = 0
    v2f a[2], b[4];
#pragma unroll
    for (int i = 0; i < 2; ++i) a[i] = *(const v2f*)(A + aoff[i]);
#pragma unroll
    for (int t = 0; t < 4; ++t) b[t] = *(const v2f*)(B + boff[t]);

    // steady state: prefetch k+4 while doing WMMAs for k
    for (int k = 4; k < K; k += 4) {
        v2f an[2], bn[4];
#pragma unroll
        for (int i = 0; i < 2; ++i) an[i] = *(const v2f*)(A + aoff[i] + k);
#pragma unroll
        for (int t = 0; t < 4; ++t) bn[t] = *(const v2f*)(B + boff[t] + k);

#pragma unroll
        for (int i = 0; i < 2; ++i)
#pragma unroll
            for (int t = 0; t < 4; ++t)
                acc[i][t] = __builtin_amdgcn_wmma_f32_16x16x4_f32(
                    false, a[i], false, b[t], (short)0, acc[i][t], false, false);

#pragma unroll
        for (int i = 0; i < 2; ++i) a[i] = an[i];
#pragma unroll
        for (int t = 0; t < 4; ++t) b[t] = bn[t];
    }

    // epilogue: last K-step
#pragma unroll
    for (int i = 0; i < 2; ++i)
#pragma unroll
        for (int t = 0; t < 4; ++t)
            acc[i][t] = __builtin_amdgcn_wmma_f32_16x16x4_f32(
                false, a[i], false, b[t], (short)0, acc[i][t], false, false);

    // D layout: lane l holds col n0+t*16+(l&15); VGPR r holds row +r+8*(l>>4)
#pragma unroll
    for (int i = 0; i < 2; ++i) {
        const int mbase = m0 + i * 16 + ((lane >> 4) << 3);
#pragma unroll
        for (int t = 0; t < 4; ++t) {
            const int n = n0 + t * 16 + mr;
            const float badd = bias ? bias[n] : 0.0f;
#pragma unroll
            for (int r = 0; r < 8; ++r) {
                Y[(size_t)(mbase + r) * N + n] = acc[i][t][r] + badd;
            }
        }
    }
}

// ---------------------------------------------------------------------------
// Y[m,n] = sum_d A[d,m] * B[d,n]        A:[D,M] B:[D,N] row-major
// Used once for P = Wk^T * Wq (tiny: 2 GFLOP). grid = (N/128, M/128).
// ---------------------------------------------------------------------------
__global__ __launch_bounds__(256) void gemm_atb(const float* __restrict__ A,
                                                const float* __restrict__ B,
                                                float* __restrict__ Y,
                                                int M, int N, int D)
{
    const int lane = threadIdx.x & 31;
    const int wave = threadIdx.x >> 5;
    const int m0   = blockIdx.y * 128 + wave * 16;
    const int n0   = blockIdx.x * 128;
    const int mr   = lane & 15;
    const int kofs = (lane >> 4) << 1;

    v8f acc[8];
#pragma unroll
    for (int t = 0; t < 8; ++t) acc[t] = (v8f){};

    for (int d = 0; d < D; d += 4) {
        v2f a;
        a.x = A[(unsigned)(d + kofs + 0) * (unsigned)M + (unsigned)(m0 + mr)];
        a.y = A[(unsigned)(d + kofs + 1) * (unsigned)M + (unsigned)(m0 + mr)];
#pragma unroll
        for (int t = 0; t < 8; ++t) {
            const unsigned n = (unsigned)(n0 + t * 16 + mr);
            v2f bfrag;
            bfrag.x = B[(unsigned)(d + kofs + 0) * (unsigned)N + n];
            bfrag.y = B[(unsigned)(d + kofs + 1) * (unsigned)N + n];
            acc[t] = __builtin_amdgcn_wmma_f32_16x16x4_f32(
                false, a, false, bfrag, (short)0, acc[t], false, false);
        }
    }

    const int mbase = m0 + ((lane >> 4) << 3);
#pragma unroll
    for (int t = 0; t < 8; ++t) {
        const int n = n0 + t * 16 + mr;
#pragma unroll
        for (int r = 0; r < 8; ++r) {
            Y[(size_t)(mbase + r) * N + n] = acc[t][r];
        }
    }
}

// ---------------------------------------------------------------------------
// Per-batch attention. One block (256 thr) per batch b:
//   scores[i,j] = U[b,i,:] . x[b,j,:]       (8x8, dots over 1024)
//   beta = softmax over i (axis=1)          sw_out = gamma*beta
//   o[j,c] = gamma * sum_i beta[i,j] H[b,i,c]     (o_out may alias U)
// ---------------------------------------------------------------------------
__global__ __launch_bounds__(256) void attn_kernel(const float* __restrict__ x,
                                                   const float* __restrict__ U,
                                                   const float* __restrict__ H,
                                                   const float* __restrict__ gamma,
                                                   float* __restrict__ sw_out,
                                                   float* __restrict__ o_out)
{
    const int b = blockIdx.x;
    const int t = threadIdx.x;
    const float* xb = x + (size_t)b * 8192;
    const float* Ub = U + (size_t)b * 8192;
    const float* Hb = H + (size_t)b * 8192;

    __shared__ float red[256];
    __shared__ float sc[64];
    __shared__ float beta[64];

    // ---- scores: 64 (i,j) pairs, 4 threads each ----
    const int pair = t >> 2;
    const int sub  = t & 3;
    const int i    = pair >> 3;
    const int j    = pair & 7;

    float s = 0.0f;
    {
        const float* urow = Ub + i * 1024;
        const float* xrow = xb + j * 1024;
        for (int c = sub; c < 1024; c += 4)
            s += urow[c] * xrow[c];
    }
    red[t] = s;
    __syncthreads();
    if (sub == 0)
        sc[pair] = red[t] + red[t + 1] + red[t + 2] + red[t + 3];
    __syncthreads();

    // ---- softmax over i, per column j (threads 0..7) ----
    if (t < 8) {
        const int jj = t;
        float m = -3.0e38f;
        for (int ii = 0; ii < 8; ++ii) m = fmaxf(m, sc[ii * 8 + jj]);
        float e[8];
        float sum = 0.0f;
        for (int ii = 0; ii < 8; ++ii) {
            e[ii] = __expf(sc[ii * 8 + jj] - m);
            sum += e[ii];
        }
        const float inv = 1.0f / sum;
        for (int ii = 0; ii < 8; ++ii) beta[ii * 8 + jj] = e[ii] * inv;
    }
    __syncthreads();   // also fences the last reads of Ub before o overwrites it

    const float g = gamma[0];
    if (t < 64) sw_out[(size_t)b * 64 + t] = g * beta[t];

    // ---- o = gamma * beta^T H  -> o_out (aliases U region, safe: same block)
    float* ob = o_out + (size_t)b * 8192;
    for (int idx = t; idx < 8192; idx += 256) {
        const int jj = idx >> 10;
        const int c  = idx & 1023;
        float a = 0.0f;
#pragma unroll
        for (int ii = 0; ii < 8; ++ii)
            a += beta[ii * 8 + jj] * Hb[ii * 1024 + c];
        ob[idx] = g * a;
    }
}

// ---------------------------------------------------------------------------
extern "C" void kernel_launch(void* const* d_in, const int* in_sizes, int n_in,
                              void* d_out, int out_size, void* d_ws, size_t ws_size,
                              hipStream_t stream)
{
    (void)in_sizes; (void)n_in; (void)out_size; (void)ws_size;

    const int B = 4096, T = 8, C = 1024;

    const float* x     = (const float*)d_in[0];   // [B,T,C]
    const float* Wq    = (const float*)d_in[1];   // [C,C]
    const float* Wk    = (const float*)d_in[2];   // [C,C]
    const float* Wv    = (const float*)d_in[3];   // [C,C]
    const float* Wfc   = (const float*)d_in[4];   // [C, T*C]
    const float* bfc   = (const float*)d_in[5];   // [C]
    const float* gamma = (const float*)d_in[6];   // [1]

    float* out_o  = (float*)d_out;                       // [B,C]
    float* out_sw = out_o + (size_t)B * C;               // [B,T,T]

    // workspace layout: P (4MB) | U/o (128MB) | H (128MB)
    char*  ws = (char*)d_ws;
    float* P  = (float*)ws;                                              // [C,C]
    float* Uo = (float*)(ws + ((size_t)C * C * 4));                      // [B, T*C]
    float* Hm = (float*)(ws + ((size_t)C * C * 4) + ((size_t)B * T * C * 4));

    const int MT = B * T;   // 32768 rows of x

    // 1) P = Wk^T * Wq   (so U = x @ P^T reproduces x @ (Wq^T Wk))
    gemm_atb<<<dim3(C / 128, C / 128), 256, 0, stream>>>(Wk, Wq, P, C, C, C);

    // 2) U = x @ P^T    [32768, 1024]
    gemm_abt<<<dim3(C / 256, MT / 64), 256, 0, stream>>>(x, P, Uo, MT, C, C, nullptr);

    // 3) H = x @ Wv^T   [32768, 1024]
    gemm_abt<<<dim3(C / 256, MT / 64), 256, 0, stream>>>(x, Wv, Hm, MT, C, C, nullptr);

    // 4) attention: scores -> softmax(axis=i) -> scaled_weights out, o over U
    attn_kernel<<<B, 256, 0, stream>>>(x, Uo, Hm, gamma, out_sw, Uo);

    // 5) out = o @ Wfc^T + bfc   [4096, 1024], K = 8192
    gemm_abt<<<dim3(C / 256, B / 64), 256, 0, stream>>>(Uo, Wfc, out_o, B, C, T * C, bfc);
}